// RHGNNLayer_70025146794671
// MI455X (gfx1250) — compile-verified
//
#include <hip/hip_runtime.h>
#include <math.h>

typedef float v2f __attribute__((ext_vector_type(2)));
typedef float v4f __attribute__((ext_vector_type(4)));
typedef float v8f __attribute__((ext_vector_type(8)));
typedef unsigned int u32x4 __attribute__((ext_vector_type(4)));
typedef int i32x4 __attribute__((ext_vector_type(4)));
typedef int i32x8 __attribute__((ext_vector_type(8)));

#define KHEADS 8
#define DHEAD 32
#define KD 256
#define DIN 128
#define NEGS 0.2f

// ---------------- utility ----------------

__global__ void zero_kernel(float* __restrict__ p, size_t n) {
    size_t i = (size_t)blockIdx.x * blockDim.x + threadIdx.x;
    size_t stride = (size_t)gridDim.x * blockDim.x;
    for (; i < n; i += stride) p[i] = 0.0f;
}

__device__ __forceinline__ unsigned fenc(float f) {
    unsigned u = __float_as_uint(f);
    return (u & 0x80000000u) ? ~u : (u | 0x80000000u);
}
__device__ __forceinline__ float fdec(unsigned u) {
    return (u & 0x80000000u) ? __uint_as_float(u & 0x7fffffffu) : __uint_as_float(~u);
}

// ---------------- WMMA fp32 GEMM: out(N x 256) = A(N x 128) @ W(128 x 256) ----------------
// The block's 128x64 weight slice is staged into LDS once via the Tensor Data Mover
// (tensor_load_to_lds, tracked by TENSORcnt); B fragments are then served from LDS.
// mode 0: plain store (projection matrices)
// mode 1: gated residual epilogue: out = sig(gate)*relu(out_prev) + (1-sig(gate))*(A@W + bias)
__global__ __launch_bounds__(256)
void gemm_wmma_f32(const float* __restrict__ A, const float* __restrict__ W,
                   float* __restrict__ out, int Nrows,
                   const float* __restrict__ bias, const float* __restrict__ gate,
                   int mode)
{
    __shared__ float wtile[DIN * 64];             // 32 KB: rows 0..127 x 64 cols (slice n0..n0+63)

    const int lane  = threadIdx.x & 31;
    const int wave  = threadIdx.x >> 5;
    const int Mtiles = (Nrows + 15) >> 4;
    const int mtile = blockIdx.x * 8 + wave;
    const int n0  = blockIdx.y * 64;
    const int hl  = lane >> 4;                    // half-wave: 0 or 1
    const int l15 = lane & 15;

    // ---- TDM: DMA W[0:128, n0:n0+64] -> LDS (issued by wave 0 only; EXEC is ignored by TDM) ----
    if (wave == 0) {
        const unsigned lds_off = (unsigned)(size_t)(void*)wtile;
        const unsigned long long gaddr = (unsigned long long)(size_t)(const void*)(W + n0);
        u32x4 g0;
        g0[0] = 1u;                                               // count=1, user mode
        g0[1] = lds_off;                                          // lds_addr
        g0[2] = (unsigned)(gaddr & 0xffffffffull);                // global_addr[95:64]
        g0[3] = (unsigned)((gaddr >> 32) & 0x01ffffffull)         // global_addr[120:96]
              | 0x80000000u;                                      // type=2 ("image")
        i32x8 g1;
        g1[0] = (2 << 16);                                        // data_size = 4 bytes
        g1[1] = (64 << 16);                                       // tensor_dim0 = 64
        g1[2] = (DIN << 16);                                      // tensor_dim1 = 128
        g1[3] = (64 << 16);                                       // tile_dim0 = 64
        g1[4] = DIN;                                              // tile_dim1 = 128, tile_dim2 unused
        g1[5] = KD;                                               // tensor_dim0_stride = 256 (low 32)
        g1[6] = 0;
        g1[7] = 0;
        i32x4 g2; g2[0] = 1; g2[1] = 1; g2[2] = 0; g2[3] = 0;     // tensor_dim2/3 = 1 (benign)
        i32x4 g3; g3[0] = 0; g3[1] = (1 << 16); g3[2] = 0; g3[3] = 0; // tensor_dim4 = 1
#if defined(__clang_major__) && (__clang_major__ >= 23)
        i32x8 g4 = {};
        __builtin_amdgcn_tensor_load_to_lds(g0, g1, g2, g3, g4, 0);
#else
        __builtin_amdgcn_tensor_load_to_lds(g0, g1, g2, g3, 0);
#endif
        __builtin_amdgcn_s_wait_tensorcnt(0);
    }
    __syncthreads();

    if (mtile >= Mtiles) return;                  // wave-uniform: EXEC stays all-ones for WMMA

    int rowA = mtile * 16 + l15;
    if (rowA >= Nrows) rowA = Nrows - 1;          // clamp (N is a multiple of 16 here anyway)
    const float* Arow = A + (size_t)rowA * DIN;

    v8f c0 = {}, c1 = {}, c2 = {}, c3 = {};

#pragma unroll 4
    for (int k0 = 0; k0 < DIN; k0 += 4) {
        const int kb = k0 + (hl << 1);            // A VGPR0 holds K{0,2}, VGPR1 K{1,3}
        v2f a = *(const v2f*)(Arow + kb);
        const float* wrow = wtile + kb * 64 + l15;
        v2f b0 = { wrow[ 0], wrow[ 64] };
        v2f b1 = { wrow[16], wrow[ 80] };
        v2f b2 = { wrow[32], wrow[ 96] };
        v2f b3 = { wrow[48], wrow[112] };
        c0 = __builtin_amdgcn_wmma_f32_16x16x4_f32(false, a, false, b0, (short)0, c0, false, false);
        c1 = __builtin_amdgcn_wmma_f32_16x16x4_f32(false, a, false, b1, (short)0, c1, false, false);
        c2 = __builtin_amdgcn_wmma_f32_16x16x4_f32(false, a, false, b2, (short)0, c2, false, false);
        c3 = __builtin_amdgcn_wmma_f32_16x16x4_f32(false, a, false, b3, (short)0, c3, false, false);
    }

    float al = 0.0f;
    if (mode == 1) al = 1.0f / (1.0f + expf(-gate[0]));

#pragma unroll
    for (int j = 0; j < 8; ++j) {
        const int row = mtile * 16 + hl * 8 + j;  // C/D: VGPR j -> rows j / j+8
        if (row >= Nrows) continue;
        const size_t base = (size_t)row * KD + n0 + l15;
        float v0 = c0[j], v1 = c1[j], v2 = c2[j], v3 = c3[j];
        if (mode == 1) {
            v0 += bias[n0 + l15 +  0];
            v1 += bias[n0 + l15 + 16];
            v2 += bias[n0 + l15 + 32];
            v3 += bias[n0 + l15 + 48];
            const float h0 = fmaxf(out[base +  0], 0.0f);
            const float h1 = fmaxf(out[base + 16], 0.0f);
            const float h2 = fmaxf(out[base + 32], 0.0f);
            const float h3 = fmaxf(out[base + 48], 0.0f);
            v0 = al * h0 + (1.0f - al) * v0;
            v1 = al * h1 + (1.0f - al) * v1;
            v2 = al * h2 + (1.0f - al) * v2;
            v3 = al * h3 + (1.0f - al) * v3;
        }
        out[base +  0] = v0;
        out[base + 16] = v1;
        out[base + 32] = v2;
        out[base + 48] = v3;
    }
}

// ---------------- small matvecs: attn = r @ Wrel (split L/R halves), ro = r @ Wupd + b ----------------
__global__ __launch_bounds__(512)
void attnro_kernel(const float* __restrict__ r, const float* __restrict__ Wrel,
                   const float* __restrict__ Wupd, const float* __restrict__ bupd,
                   float* __restrict__ attnL, float* __restrict__ attnR,
                   float* __restrict__ ro)
{
    __shared__ float rs[64];
    const int j = threadIdx.x;
    if (j < 64) rs[j] = r[j];
    __syncthreads();
    float acc = 0.0f;
#pragma unroll 8
    for (int t = 0; t < 64; ++t) acc += rs[t] * Wrel[t * 512 + j];
    const int k = j >> 6, c = j & 63;
    if (c < DHEAD) attnL[k * DHEAD + c] = acc;
    else           attnR[k * DHEAD + (c - DHEAD)] = acc;
    if (j < KD) {
        float a2 = 0.0f;
#pragma unroll 8
        for (int t = 0; t < 64; ++t) a2 += rs[t] * Wupd[t * KD + j];
        ro[j] = a2 + bupd[j];
    }
}

// ---------------- per-(node,head) dot: out[n,k] = sum_d P[n,k*32+d]*att[k*32+d] ----------------
__global__ __launch_bounds__(256)
void dot32_kernel(const float* __restrict__ P, const float* __restrict__ att,
                  float* __restrict__ out, int Nrows)
{
    const int i = blockIdx.x * blockDim.x + threadIdx.x;
    if (i >= Nrows * KHEADS) return;
    const int n = i >> 3, k = i & 7;
    const float* p = P + (size_t)n * KD + k * DHEAD;
    const float* a = att + k * DHEAD;
    float s = 0.0f;
#pragma unroll
    for (int d = 0; d < DHEAD; d += 4) {
        v4f pv = *(const v4f*)(p + d);
        v4f av = *(const v4f*)(a + d);
        s += pv.x * av.x + pv.y * av.y + pv.z * av.z + pv.w * av.w;
    }
    out[i] = s;
}

// ---------------- edge pass ----------------
__global__ __launch_bounds__(256)
void edge_emax_kernel(const int* __restrict__ src, const int* __restrict__ dst,
                      const float* __restrict__ el, const float* __restrict__ er,
                      float* __restrict__ ebuf, unsigned* __restrict__ menc, int E)
{
    const int i = blockIdx.x * blockDim.x + threadIdx.x;
    if (i >= E * KHEADS) return;
    const int e = i >> 3, k = i & 7;
    const int s = src[e], d = dst[e];
    float v = el[s * KHEADS + k] + er[d * KHEADS + k];
    v = v > 0.0f ? v : NEGS * v;
    ebuf[i] = v;
    atomicMax(&menc[d * KHEADS + k], fenc(v));
}

__global__ __launch_bounds__(256)
void edge_sum_kernel(const int* __restrict__ dst, const float* __restrict__ ebuf,
                     const unsigned* __restrict__ menc, float* __restrict__ sbuf, int E)
{
    const int i = blockIdx.x * blockDim.x + threadIdx.x;
    if (i >= E * KHEADS) return;
    const int d = dst[i >> 3], k = i & 7;
    const float m = fdec(menc[d * KHEADS + k]);
    atomicAdd(&sbuf[d * KHEADS + k], expf(ebuf[i] - m));
}

// one wave per edge: lane covers one d-column per head slice (coalesced gather + atomic add)
__global__ __launch_bounds__(256)
void edge_acc_kernel(const int* __restrict__ src, const int* __restrict__ dst,
                     const float* __restrict__ ebuf, const unsigned* __restrict__ menc,
                     const float* __restrict__ sbuf, const float* __restrict__ P,
                     float* __restrict__ h, int E)
{
    const int t = blockIdx.x * blockDim.x + threadIdx.x;
    const int e = t >> 5, lane = t & 31;
    if (e >= E) return;
    const int s = src[e], d = dst[e];
    float aa[KHEADS];
#pragma unroll
    for (int k = 0; k < KHEADS; ++k) {
        const float m  = fdec(menc[d * KHEADS + k]);
        const float ss = sbuf[d * KHEADS + k];
        aa[k] = expf(ebuf[e * KHEADS + k] - m) / ss;
    }
    const float* ps = P + (size_t)s * KD + lane;
    float* hd = h + (size_t)d * KD + lane;
#pragma unroll
    for (int j = 0; j < KHEADS; ++j)
        atomicAdd(hd + j * DHEAD, ps[j * DHEAD] * aa[j]);
}

// ---------------- cross-relation fusion over the 3 item relations (in place) ----------------
__global__ __launch_bounds__(256)
void crossing_kernel(float* __restrict__ h0, float* __restrict__ h2, float* __restrict__ h3,
                     const float* __restrict__ a0, const float* __restrict__ a1,
                     const float* __restrict__ a2, int Nrows)
{
    const int i = blockIdx.x * blockDim.x + threadIdx.x;
    if (i >= Nrows * KHEADS) return;
    const int n = i >> 3, k = i & 7;
    const size_t base = (size_t)n * KD + k * DHEAD;

    float f[3][DHEAD];
#pragma unroll
    for (int d = 0; d < DHEAD; d += 4) {
        *(v4f*)&f[0][d] = *(const v4f*)(h0 + base + d);
        *(v4f*)&f[1][d] = *(const v4f*)(h2 + base + d);
        *(v4f*)&f[2][d] = *(const v4f*)(h3 + base + d);
    }
    const float* atts[3] = { a0, a1, a2 };
    float* outs[3] = { h0, h2, h3 };

    for (int X = 0; X < 3; ++X) {
        const float* av = atts[X] + k * DHEAD;
        float sc0 = 0.0f, sc1 = 0.0f, sc2 = 0.0f;
#pragma unroll
        for (int d = 0; d < DHEAD; ++d) {
            const float a = av[d];
            sc0 += a * f[0][d];
            sc1 += a * f[1][d];
            sc2 += a * f[2][d];
        }
        sc0 = sc0 > 0.0f ? sc0 : NEGS * sc0;
        sc1 = sc1 > 0.0f ? sc1 : NEGS * sc1;
        sc2 = sc2 > 0.0f ? sc2 : NEGS * sc2;
        const float m = fmaxf(sc0, fmaxf(sc1, sc2));
        float w0 = expf(sc0 - m), w1 = expf(sc1 - m), w2 = expf(sc2 - m);
        const float inv = 1.0f / (w0 + w1 + w2);
        w0 *= inv; w1 *= inv; w2 *= inv;
        float* o = outs[X] + base;
#pragma unroll
        for (int d = 0; d < DHEAD; ++d)
            o[d] = w0 * f[0][d] + w1 * f[1][d] + w2 * f[2][d];
    }
}

// ---------------- host ----------------
extern "C" void kernel_launch(void* const* d_in, const int* in_sizes, int n_in,
                              void* d_out, int out_size, void* d_ws, size_t ws_size,
                              hipStream_t stream)
{
    const float* feat[4]  = { (const float*)d_in[0], (const float*)d_in[1],
                              (const float*)d_in[2], (const float*)d_in[3] };
    const float* rvec[4]  = { (const float*)d_in[4], (const float*)d_in[5],
                              (const float*)d_in[6], (const float*)d_in[7] };
    const float* W_user   = (const float*)d_in[8];
    const float* W_item   = (const float*)d_in[9];
    const float* Wrel[4]  = { (const float*)d_in[10], (const float*)d_in[11],
                              (const float*)d_in[12], (const float*)d_in[13] };
    const float* catt[4]  = { (const float*)d_in[14], (const float*)d_in[15],
                              (const float*)d_in[16], (const float*)d_in[17] };
    const float* Wres_user = (const float*)d_in[18];
    const float* bres_user = (const float*)d_in[19];
    const float* Wres_item = (const float*)d_in[20];
    const float* bres_item = (const float*)d_in[21];
    const float* a_user    = (const float*)d_in[22];
    const float* a_item    = (const float*)d_in[23];
    const float* Wupd[4]  = { (const float*)d_in[24], (const float*)d_in[26],
                              (const float*)d_in[28], (const float*)d_in[30] };
    const float* bupd[4]  = { (const float*)d_in[25], (const float*)d_in[27],
                              (const float*)d_in[29], (const float*)d_in[31] };
    const int* srcI[4]    = { (const int*)d_in[32], (const int*)d_in[34],
                              (const int*)d_in[36], (const int*)d_in[38] };
    const int* dstI[4]    = { (const int*)d_in[33], (const int*)d_in[35],
                              (const int*)d_in[37], (const int*)d_in[39] };

    const int N = in_sizes[0] / DIN;
    const int Ecnt[4] = { in_sizes[32], in_sizes[34], in_sizes[36], in_sizes[38] };

    // ---- workspace layout ----
    float* ws = (float*)d_ws;
    const size_t NN = (size_t)N * KD;
    float* Pbuf[4] = { ws, ws + NN, ws + 2 * NN, ws + 3 * NN };
    float* attL = ws + 4 * NN;                       // 4 * 256
    float* attR = attL + 4 * KD;                     // 4 * 256
    float* elb  = attR + 4 * KD;                     // 4 * N*8
    float* erb  = elb + 4 * (size_t)N * KHEADS;      // 4 * N*8
    unsigned* menc = (unsigned*)(erb + 4 * (size_t)N * KHEADS);   // N*8
    float* sbuf = (float*)menc + (size_t)N * KHEADS;              // N*8
    float* ebuf = sbuf + (size_t)N * KHEADS;                      // Emax*8

    // ---- output layout: 4 h/out slots then 4 relation-update vectors ----
    float* outp = (float*)d_out;
    float* hslot[4] = { outp, outp + NN, outp + 2 * NN, outp + 3 * NN };
    float* rout = outp + 4 * NN;

    // relation r: fs = Pbuf[fsIdx[r]], fd = Pbuf[fdIdx[r]]
    // Pbuf0 = feat_iu@W_user, Pbuf1 = feat_ui@W_item, Pbuf2 = feat_iirev@W_item, Pbuf3 = feat_ii@W_item
    const int fsIdx[4] = { 0, 1, 2, 3 };
    const int fdIdx[4] = { 1, 0, 3, 2 };

    const int Mtiles = (N + 15) >> 4;
    dim3 ggrid((Mtiles + 7) / 8, 4);

    // 1) zero the 4 h slots (edge pass accumulates into them)
    zero_kernel<<<2048, 256, 0, stream>>>(outp, 4 * NN);

    // 2) relation attention vectors + relation feature updates
    for (int r = 0; r < 4; ++r)
        attnro_kernel<<<1, 512, 0, stream>>>(rvec[r], Wrel[r], Wupd[r], bupd[r],
                                             attL + r * KD, attR + r * KD, rout + r * KD);

    // 3) node projections (WMMA GEMMs, TDM-staged weights)
    gemm_wmma_f32<<<ggrid, 256, 0, stream>>>(feat[1], W_user, Pbuf[0], N, nullptr, nullptr, 0);
    gemm_wmma_f32<<<ggrid, 256, 0, stream>>>(feat[0], W_item, Pbuf[1], N, nullptr, nullptr, 0);
    gemm_wmma_f32<<<ggrid, 256, 0, stream>>>(feat[3], W_item, Pbuf[2], N, nullptr, nullptr, 0);
    gemm_wmma_f32<<<ggrid, 256, 0, stream>>>(feat[2], W_item, Pbuf[3], N, nullptr, nullptr, 0);

    // 4) per-(node,head) attention logits el / er
    const int dotBlocks = (N * KHEADS + 255) / 256;
    for (int r = 0; r < 4; ++r) {
        dot32_kernel<<<dotBlocks, 256, 0, stream>>>(Pbuf[fsIdx[r]], attL + r * KD,
                                                    elb + (size_t)r * N * KHEADS, N);
        dot32_kernel<<<dotBlocks, 256, 0, stream>>>(Pbuf[fdIdx[r]], attR + r * KD,
                                                    erb + (size_t)r * N * KHEADS, N);
    }

    // 5) edge softmax + message accumulation per relation
    for (int r = 0; r < 4; ++r) {
        const int E = Ecnt[r];
        zero_kernel<<<1024, 256, 0, stream>>>((float*)menc, 2 * (size_t)N * KHEADS); // menc + sbuf
        const int ekBlocks = (E * KHEADS + 255) / 256;
        edge_emax_kernel<<<ekBlocks, 256, 0, stream>>>(srcI[r], dstI[r],
            elb + (size_t)r * N * KHEADS, erb + (size_t)r * N * KHEADS, ebuf, menc, E);
        edge_sum_kernel<<<ekBlocks, 256, 0, stream>>>(dstI[r], ebuf, menc, sbuf, E);
        const int accBlocks = (E * 32 + 255) / 256;
        edge_acc_kernel<<<accBlocks, 256, 0, stream>>>(srcI[r], dstI[r], ebuf, menc, sbuf,
                                                       Pbuf[fsIdx[r]], hslot[r], E);
    }

    // 6) gated residual (fused GEMM epilogue): h = sig(a)*relu(h) + (1-sig(a))*(feat@Wres+b)
    gemm_wmma_f32<<<ggrid, 256, 0, stream>>>(feat[0], Wres_item, hslot[0], N, bres_item, a_item, 1);
    gemm_wmma_f32<<<ggrid, 256, 0, stream>>>(feat[1], Wres_user, hslot[1], N, bres_user, a_user, 1);
    gemm_wmma_f32<<<ggrid, 256, 0, stream>>>(feat[2], Wres_item, hslot[2], N, bres_item, a_item, 1);
    gemm_wmma_f32<<<ggrid, 256, 0, stream>>>(feat[3], Wres_item, hslot[3], N, bres_item, a_item, 1);

    // 7) cross-relation fusion for item relations (slot1 = out_iu is identity, already final)
    crossing_kernel<<<dotBlocks, 256, 0, stream>>>(hslot[0], hslot[2], hslot[3],
                                                   catt[0], catt[2], catt[3], N);
}